// MaskerImage_39651138077505
// MI455X (gfx1250) — compile-verified
//
#include <hip/hip_runtime.h>
#include <cstdint>

// ---------------------------------------------------------------------------
// MaskerImage for MI455X (gfx1250, wave32).
// Memory-bound: ~470-540MB of traffic -> ~20us floor @ 23.3 TB/s. No matmul
// structure => no WMMA; the CDNA5 paths used here are b128 vector VMEM,
// non-temporal hints (protect 192MB L2 for the mask re-read), LDS-atomic
// histograms for exact per-sample top-k, and async global->LDS staging
// (ASYNCcnt) to overlap threefry/erfinv VALU work with the mask fetch.
// ---------------------------------------------------------------------------

typedef float v4f __attribute__((ext_vector_type(4)));

#define HW      262144u    // 512*512
#define CHW     786432u    // 3*HW
#define NX      50331648u  // 64*CHW
#define NM      16777216u  // 64*HW
#define HALF_M  8388608u   // NM/2   (32 samples)
#define HALF_N  25165824u  // NX/2   (= 32*CHW)
#define LEN_KEEP 183500u   // int(262144*0.7)

struct KeyPair { unsigned a, b; };

__host__ __device__ constexpr unsigned crotl(unsigned x, int r) {
  return (x << r) | (x >> (32 - r));
}

// JAX threefry2x32 (20 rounds), bit-exact structure.
__host__ __device__ constexpr KeyPair threefry2x32(unsigned k0, unsigned k1,
                                                   unsigned c0, unsigned c1) {
  unsigned ks2 = k0 ^ k1 ^ 0x1BD11BDAu;
  unsigned x0 = c0 + k0, x1 = c1 + k1;
  const int RA[4] = {13, 15, 26, 6};
  const int RB[4] = {17, 29, 16, 24};
  for (int i = 0; i < 4; i++) { x0 += x1; x1 = crotl(x1, RA[i]); x1 ^= x0; }
  x0 += k1;  x1 += ks2 + 1u;
  for (int i = 0; i < 4; i++) { x0 += x1; x1 = crotl(x1, RB[i]); x1 ^= x0; }
  x0 += ks2; x1 += k0 + 2u;
  for (int i = 0; i < 4; i++) { x0 += x1; x1 = crotl(x1, RA[i]); x1 ^= x0; }
  x0 += k0;  x1 += k1 + 3u;
  for (int i = 0; i < 4; i++) { x0 += x1; x1 = crotl(x1, RB[i]); x1 ^= x0; }
  x0 += k1;  x1 += ks2 + 4u;
  for (int i = 0; i < 4; i++) { x0 += x1; x1 = crotl(x1, RA[i]); x1 ^= x0; }
  x0 += ks2; x1 += k0 + 5u;
  return {x0, x1};
}

// jax.random.key(42) = (0,42); split(): keys = threefry(key,[0,1,2,3]) pairs (0,2),(1,3)
// k_mask = (o0(0,2), o0(1,3));  k_noise = (o1(0,2), o1(1,3))
constexpr KeyPair SP02 = threefry2x32(0u, 42u, 0u, 2u);
constexpr KeyPair SP13 = threefry2x32(0u, 42u, 1u, 3u);
constexpr unsigned KM0 = SP02.a, KM1 = SP13.a;
constexpr unsigned KN0 = SP02.b, KN1 = SP13.b;

// noise = 0.1*sqrt(2)*erfinv(uniform(lo,1)) + 1, lo = nextafterf(-1,0)
__device__ __forceinline__ float noise_from_bits(unsigned bits) {
  float f = __uint_as_float((bits >> 9) | 0x3f800000u) - 1.0f; // [0,1), 23 bits
  const float LO = -0.99999994f;
  float u = f * (1.0f - LO) + LO;
  u = fmaxf(u, LO);
  return fmaf(1.41421356237309515f * erfinvf(u), 0.1f, 1.0f);
}

// ---- Pass 1: per-sample 2048-bin histogram of top-11 bits of the sort key.
// 16 threefry evals per thread so the LDS zero/flush is amortized 16x.
__global__ void k_hist1(unsigned* __restrict__ gh) {
  const unsigned t = threadIdx.x;
  const unsigned base = blockIdx.x * 4096u;   // 4096 pair-indices per block
  const unsigned sA = base / HW, sB = sA + 32u;  // one sample per block (4096 | HW)
  __shared__ unsigned hs[4096];
  for (unsigned k = t; k < 4096u; k += 256u) hs[k] = 0u;
  __syncthreads();
  for (unsigned k = 0; k < 16u; k++) {
    unsigned i = base + k * 256u + t;         // partner = i + HALF_M
    KeyPair r = threefry2x32(KM0, KM1, i, i + HALF_M);
    atomicAdd(&hs[r.a >> 21], 1u);
    atomicAdd(&hs[2048u + (r.b >> 21)], 1u);
  }
  __syncthreads();
  for (unsigned k = t; k < 4096u; k += 256u) {
    unsigned c = hs[k];
    if (c) atomicAdd(&gh[((k < 2048u) ? sA : sB) * 2048u + (k & 2047u)], c);
  }
}

// ---- Pass 2: per-sample scan of 2048 bins -> coarse threshold bin + remainder ----
__global__ void k_sel1(const unsigned* __restrict__ h, unsigned* __restrict__ sel) {
  const unsigned s = blockIdx.x, t = threadIdx.x;
  const unsigned* hb = h + s * 2048u;
  unsigned loc[8]; unsigned psum = 0u;
  for (int k = 0; k < 8; k++) { loc[k] = hb[t * 8u + k]; psum += loc[k]; }
  __shared__ unsigned sc[256];
  sc[t] = psum; __syncthreads();
  for (int off = 1; off < 256; off <<= 1) {
    unsigned v = sc[t]; unsigned a = (t >= (unsigned)off) ? sc[t - off] : 0u;
    __syncthreads(); sc[t] = v + a; __syncthreads();
  }
  unsigned c = sc[t] - psum;   // exclusive prefix
  for (int k = 0; k < 8; k++) {
    unsigned hk = loc[k];
    if (c < LEN_KEEP && LEN_KEEP <= c + hk) {
      sel[2u * s] = t * 8u + k;            // coarse bin T
      sel[2u * s + 1u] = LEN_KEEP - c;     // k_rem in [1, h[T]]
    }
    c += hk;
  }
}

// ---- Pass 3: per-sample 4096-bin histogram of low-12 bits, restricted to bin T ----
__global__ void k_hist2(unsigned* __restrict__ gh2, const unsigned* __restrict__ sel1) {
  const unsigned t = threadIdx.x;
  const unsigned base = blockIdx.x * 4096u;
  const unsigned sA = base / HW, sB = sA + 32u;
  const unsigned TA = sel1[2u * sA], TB = sel1[2u * sB];
  __shared__ unsigned hs[8192];
  for (unsigned k = t; k < 8192u; k += 256u) hs[k] = 0u;
  __syncthreads();
  for (unsigned k = 0; k < 16u; k++) {
    unsigned i = base + k * 256u + t;
    KeyPair r = threefry2x32(KM0, KM1, i, i + HALF_M);
    unsigned vA = r.a >> 9, vB = r.b >> 9;
    if ((vA >> 12) == TA) atomicAdd(&hs[vA & 0xFFFu], 1u);
    if ((vB >> 12) == TB) atomicAdd(&hs[4096u + (vB & 0xFFFu)], 1u);
  }
  __syncthreads();
  for (unsigned k = t; k < 8192u; k += 256u) {
    unsigned c = hs[k];
    if (c) atomicAdd(&gh2[((k < 4096u) ? sA : sB) * 4096u + (k & 4095u)], c);
  }
}

// ---- Pass 4: scan 4096 fine bins -> exact 23-bit threshold + tie remainder ----
__global__ void k_sel2(const unsigned* __restrict__ h2,
                       const unsigned* __restrict__ sel1,
                       unsigned* __restrict__ sel2) {
  const unsigned s = blockIdx.x, t = threadIdx.x;
  const unsigned* hb = h2 + s * 4096u;
  const unsigned T1 = sel1[2u * s];
  const unsigned K  = sel1[2u * s + 1u];
  unsigned loc[16]; unsigned psum = 0u;
  for (int k = 0; k < 16; k++) { loc[k] = hb[t * 16u + k]; psum += loc[k]; }
  __shared__ unsigned sc[256];
  sc[t] = psum; __syncthreads();
  for (int off = 1; off < 256; off <<= 1) {
    unsigned v = sc[t]; unsigned a = (t >= (unsigned)off) ? sc[t - off] : 0u;
    __syncthreads(); sc[t] = v + a; __syncthreads();
  }
  unsigned c = sc[t] - psum;
  for (int k = 0; k < 16; k++) {
    unsigned hk = loc[k];
    if (c < K && K <= c + hk) {
      sel2[2u * s] = (T1 << 12) | (t * 16u + k);  // full 23-bit threshold value
      sel2[2u * s + 1u] = K - c;                  // k_rem2: #ties to keep (>=1)
    }
    c += hk;
  }
}

// ---- Pass 5: write mask (RT stores: 67MB stays resident in 192MB L2 for k_final).
// Pixels strictly below threshold -> 1, above -> 0. Pixels equal to the
// threshold value (expected ~1 per sample) are written 0 and appended to a
// per-sample global list; k_tiefix patches the kept ones to 1. This folds the
// tie scan into the mask pass, removing an entire 8.4M-eval RNG pass.
__global__ void k_mask(const unsigned* __restrict__ sel2,
                       unsigned* __restrict__ tcnt,
                       unsigned* __restrict__ tidx,
                       float* __restrict__ mask) {
  const unsigned t = threadIdx.x;
  const unsigned i0 = (blockIdx.x * 256u + t) * 4u;
  const unsigned sA = i0 / HW, p0 = i0 % HW, sB = sA + 32u;
  const unsigned thrA = sel2[2u * sA], thrB = sel2[2u * sB];
  v4f ma, mb;
  for (int l = 0; l < 4; l++) {
    KeyPair r = threefry2x32(KM0, KM1, i0 + l, i0 + l + HALF_M);
    unsigned vA = r.a >> 9, vB = r.b >> 9;
    ma[l] = (vA < thrA) ? 1.0f : 0.0f;
    mb[l] = (vB < thrB) ? 1.0f : 0.0f;
    if (vA == thrA) {
      unsigned pos = atomicAdd(&tcnt[sA], 1u);
      if (pos < 1024u) tidx[sA * 1024u + pos] = p0 + l;
    }
    if (vB == thrB) {
      unsigned pos = atomicAdd(&tcnt[sB], 1u);
      if (pos < 1024u) tidx[sB * 1024u + pos] = p0 + l;
    }
  }
  *(v4f*)(mask + (size_t)sA * HW + p0) = ma;
  *(v4f*)(mask + (size_t)sB * HW + p0) = mb;
}

// ---- Pass 6: stable-argsort tie-break. Keep the k_rem2 lowest pixel indices
// among the (tiny) tie list; patch mask=1 there. Non-destructive k-th-smallest
// (pixel indices are unique per sample).
__global__ void k_tiefix(const unsigned* __restrict__ sel2,
                         const unsigned* __restrict__ tcnt,
                         const unsigned* __restrict__ tidx,
                         float* __restrict__ mask) {
  const unsigned s = blockIdx.x, t = threadIdx.x;
  const unsigned need = sel2[2u * s + 1u];
  unsigned n = tcnt[s]; n = n < 1024u ? n : 1024u;
  const unsigned* lst = tidx + s * 1024u;
  __shared__ unsigned sbound;
  if (t == 0u) {
    unsigned bound;
    if (need >= n) {                       // keep all ties
      unsigned mx = 0u;
      for (unsigned k = 0; k < n; k++) { unsigned v = lst[k]; mx = v > mx ? v : mx; }
      bound = mx + 1u;
    } else {                               // need-th smallest, non-destructive
      unsigned prev = 0u; bool first = true;
      for (unsigned r2 = 0; r2 < need; r2++) {
        unsigned mn = 0xFFFFFFFFu;
        for (unsigned k = 0; k < n; k++) {
          unsigned v = lst[k];
          if ((first || v > prev) && v < mn) mn = v;
        }
        prev = mn; first = false;
      }
      bound = prev + 1u;
    }
    sbound = bound;
  }
  __syncthreads();
  const unsigned bound = sbound;
  for (unsigned k = t; k < n; k += 256u) {
    unsigned p = lst[k];
    if (p < bound) mask[(size_t)s * HW + p] = 1.0f;
  }
}

// ---- Pass 7 (fused): out = x * mask * noise.
// Async-stage mask tiles to LDS (ASYNCcnt) while each thread runs 12 threefry
// evals + 24 erfinv; NT loads for x and NT stores for out keep the 400MB
// streams out of L2 so the mask read hits L2.
__global__ void k_final(const float* __restrict__ x,
                        const float* __restrict__ mask,
                        float* __restrict__ out) {
  const unsigned t = threadIdx.x;
  const unsigned bp = blockIdx.x >> 8;        // sample pair: (bp, bp+32)
  const unsigned tile = blockIdx.x & 255u;
  const unsigned p0 = tile * 1024u + t * 4u;  // pixel base for this thread

  __shared__ v4f shm[512];                    // [0..255]=mask(bp), [256..511]=mask(bp+32)
  unsigned long long gA = (unsigned long long)(const void*)(mask + (size_t)bp * HW + p0);
  unsigned long long gB = (unsigned long long)(const void*)(mask + (size_t)(bp + 32u) * HW + p0);
  unsigned la = (unsigned)(uintptr_t)&shm[t];
  unsigned lb = (unsigned)(uintptr_t)&shm[256u + t];
  asm volatile("global_load_async_to_lds_b128 %0, %1, off" :: "v"(la), "v"(gA) : "memory");
  asm volatile("global_load_async_to_lds_b128 %0, %1, off" :: "v"(lb), "v"(gB) : "memory");

  v4f xa[3], xb[3];
  float nzA[3][4], nzB[3][4];
  for (int c = 0; c < 3; c++) {
    unsigned e = bp * CHW + (unsigned)c * HW + p0;
    xa[c] = __builtin_nontemporal_load((const v4f*)(x + e));
    xb[c] = __builtin_nontemporal_load((const v4f*)(x + e + HALF_N));
    for (int l = 0; l < 4; l++) {
      KeyPair r = threefry2x32(KN0, KN1, e + l, e + l + HALF_N);
      nzA[c][l] = noise_from_bits(r.a);
      nzB[c][l] = noise_from_bits(r.b);
    }
  }

  asm volatile("s_wait_asynccnt 0x0" ::: "memory");
  v4f ma = shm[t];
  v4f mb = shm[256u + t];

  for (int c = 0; c < 3; c++) {
    unsigned e = bp * CHW + (unsigned)c * HW + p0;
    v4f oa, ob;
    for (int l = 0; l < 4; l++) {
      oa[l] = xa[c][l] * ma[l] * nzA[c][l];
      ob[l] = xb[c][l] * mb[l] * nzB[c][l];
    }
    __builtin_nontemporal_store(oa, (v4f*)(out + e));
    __builtin_nontemporal_store(ob, (v4f*)(out + e + HALF_N));
  }
}

// ---------------------------------------------------------------------------
extern "C" void kernel_launch(void* const* d_in, const int* in_sizes, int n_in,
                              void* d_out, int out_size, void* d_ws, size_t ws_size,
                              hipStream_t stream) {
  (void)in_sizes; (void)n_in; (void)out_size; (void)ws_size;
  const float* x = (const float*)d_in[0];
  float* out = (float*)d_out;
  float* mask_out = out + (size_t)NX;

  uint8_t* ws = (uint8_t*)d_ws;
  unsigned* h1   = (unsigned*)(ws);               // 64*2048*4 = 512KB
  unsigned* h2   = (unsigned*)(ws + 524288);      // 64*4096*4 = 1MB
  unsigned* sel1 = (unsigned*)(ws + 1572864);     // 64*2 u32
  unsigned* sel2 = (unsigned*)(ws + 1573376);     // 64*2 u32
  unsigned* tcnt = (unsigned*)(ws + 1573888);     // 64 u32
  unsigned* tidx = (unsigned*)(ws + 1574144);     // 64*1024 u32 = 256KB
  const size_t WS_NEED = 1574144u + 262144u;      // ~1.8MB

  hipMemsetAsync(d_ws, 0, WS_NEED, stream);       // histograms/counters start at 0

  k_hist1 <<<dim3(HALF_M / 4096u), dim3(256), 0, stream>>>(h1);
  k_sel1  <<<dim3(64),             dim3(256), 0, stream>>>(h1, sel1);
  k_hist2 <<<dim3(HALF_M / 4096u), dim3(256), 0, stream>>>(h2, sel1);
  k_sel2  <<<dim3(64),             dim3(256), 0, stream>>>(h2, sel1, sel2);
  k_mask  <<<dim3(HALF_M / 1024u), dim3(256), 0, stream>>>(sel2, tcnt, tidx, mask_out);
  k_tiefix<<<dim3(64),             dim3(256), 0, stream>>>(sel2, tcnt, tidx, mask_out);
  k_final <<<dim3(32u * 256u),     dim3(256), 0, stream>>>(x, mask_out, out);
}